// GraphVAE_55705725829675
// MI455X (gfx1250) — compile-verified
//
#include <hip/hip_runtime.h>
#include <hip/hip_bf16.h>

// ---------------------------------------------------------------------------
// Graph U-Net VAE forward for MI455X (gfx1250, wave32).
// Dense GEMMs run through V_WMMA_F32_16X16X4_F32.
//  - K4<=32: templated kernel, B fragments held in registers per column tile,
//    fully unrolled WMMA chain, zero LDS traffic in the hot loop.
//  - K4>32 : generic software-pipelined kernel (pair-interleaved W in LDS).
// Net is HBM/L2-bandwidth bound at C<=32 -> fp32 WMMA matches reference math.
// ---------------------------------------------------------------------------

#define NET   7
#define C8C   24
#define C7C   32
#define C6C   32
#define LATC  3
#define NCLSC 21
#define NHC   32

#define N6C 8192
#define S6C 6144
#define L6C 2048
#define N7C 49152
#define S7C 36864
#define L7C 12288
#define N8C 294912
#define G6C 8192
#define G7C 51200
#define G8C 309248

typedef __attribute__((ext_vector_type(2))) float v2f;
typedef __attribute__((ext_vector_type(8))) float v8f;

__device__ __forceinline__ float gelu_f(float x) {
    const float k0 = 0.7978845608028654f;  // sqrt(2/pi)
    const float k1 = 0.044715f;
    return 0.5f * x * (1.0f + tanhf(k0 * (x + k1 * x * x * x)));
}

// -------------------------------- utility kernels --------------------------

__global__ __launch_bounds__(256) void zero_kernel(float* __restrict__ p, long long n) {
    long long i = (long long)blockIdx.x * blockDim.x + threadIdx.x;
    long long st = (long long)gridDim.x * blockDim.x;
    for (; i < n; i += st) p[i] = 0.0f;
}

// copy last tailRows rows (C cols) of src (srcRows x C) into dst
__global__ __launch_bounds__(256) void copy_tail_kernel(const float* __restrict__ src,
                                                        long long srcRows, int C,
                                                        long long tailRows,
                                                        float* __restrict__ dst) {
    long long total = tailRows * C;
    long long base = (srcRows - tailRows) * (long long)C;
    long long i = (long long)blockIdx.x * blockDim.x + threadIdx.x;
    long long st = (long long)gridDim.x * blockDim.x;
    for (; i < total; i += st) dst[i] = src[base + i];
}

// ------------------------- per-channel stats (group_norm) ------------------
// All group sizes degenerate to 1 channel -> batch-norm style column stats.
// stats[0..63] = sum per channel, stats[64..127] = sumsq per channel.

__global__ __launch_bounds__(256) void colstats_kernel(const float* __restrict__ X,
                                                       long long n, int C,
                                                       float* __restrict__ stats) {
    __shared__ float sh[128];
    int tid = threadIdx.x;
    if (tid < 128) sh[tid] = 0.0f;
    __syncthreads();
    int perB = 256 / C;             // rows covered by one block step
    int c  = tid % C;
    int rl = tid / C;
    if (rl < perB) {
        float s = 0.0f, ss = 0.0f;
        long long rstep = (long long)gridDim.x * perB;
        for (long long r = (long long)blockIdx.x * perB + rl; r < n; r += rstep) {
            float v = X[r * C + c];
            s += v;
            ss += v * v;
        }
        atomicAdd(&sh[c], s);          // ds_add_f32
        atomicAdd(&sh[64 + c], ss);
    }
    __syncthreads();
    if (tid < C) {
        atomicAdd(&stats[tid], sh[tid]);          // global_atomic_add_f32
        atomicAdd(&stats[64 + tid], sh[64 + tid]);
    }
}

__global__ __launch_bounds__(256) void gn_affine_kernel(const float* __restrict__ X,
                                                        const float* __restrict__ stats,
                                                        const float* __restrict__ w,
                                                        const float* __restrict__ b,
                                                        float* __restrict__ Y,
                                                        long long n, int C, int do_gelu) {
    __shared__ float scale[64], shift[64];
    int tid = threadIdx.x;
    if (tid < C) {
        float inv_n = 1.0f / (float)n;
        float mean = stats[tid] * inv_n;
        float var = stats[64 + tid] * inv_n - mean * mean;
        float sc = rsqrtf(var + 1e-5f) * w[tid];
        scale[tid] = sc;
        shift[tid] = b[tid] - mean * sc;
    }
    __syncthreads();
    long long total = n * (long long)C;
    long long i = (long long)blockIdx.x * blockDim.x + tid;
    long long st = (long long)gridDim.x * blockDim.x;
    int c = (int)(i % C);
    int cstep = (int)(st % C);
    for (; i < total; i += st) {
        float v = X[i] * scale[c] + shift[c];
        Y[i] = do_gelu ? gelu_f(v) : v;
        c += cstep;
        if (c >= C) c -= C;
    }
}

// ------------------------------- WMMA GEMM ---------------------------------
// Y[n,N] = act( X[n,K] @ W[K,N] + bias (+ residual) )
// A loads are unconditional via row clamping (OOB rows computed, never stored;
// small row over-read is safe: all GEMM inputs are interior workspace).

#define GEMM_WAVES 8

// ---- small-K path: B operand held in registers, fully unrolled WMMA chain.
template <int KQ>   // number of k-quads; K4 = 4*KQ <= 32
__global__ __launch_bounds__(256) void gemm16_reg_kernel(const float* __restrict__ X,
                                                         const float* __restrict__ W,
                                                         const float* __restrict__ bias,
                                                         const float* __restrict__ residual,
                                                         float* __restrict__ Y,
                                                         int n, int K, int N, int fuse_gelu) {
    extern __shared__ float smem[];  // K4 * N16 floats, pair-interleaved
    const int K4 = KQ * 4;
    int N16 = (N + 15) & ~15;
    int tid = threadIdx.x;
    int npairs = (K4 >> 1) * N16;
    for (int i = tid; i < npairs; i += 256) {
        int kp = i / N16;
        int col = i - kp * N16;
        int k = kp << 1;
        float w0 = (k < K && col < N) ? W[k * N + col] : 0.0f;
        float w1 = ((k + 1) < K && col < N) ? W[(k + 1) * N + col] : 0.0f;
        smem[i * 2] = w0;
        smem[i * 2 + 1] = w1;
    }
    __syncthreads();

    int wave = tid >> 5;
    int lane = tid & 31;
    int hi = lane >> 4;        // A 16x4 layout: lanes 0-15 -> K=k0,k0+1 ; 16-31 -> k0+2,k0+3
    int l15 = lane & 15;
    int ntiles = N16 >> 4;
    long long tileStride = (long long)gridDim.x * (16 * GEMM_WAVES);

    for (int t = 0; t < ntiles; ++t) {
        int col = (t << 4) + l15;
        v2f breg[KQ];
#pragma unroll
        for (int j = 0; j < KQ; ++j) {          // k = 4j + 2*hi -> pair index 2j + hi
            const float* wp = smem + ((((j << 1) + hi) * N16 + col) << 1);
            breg[j].x = wp[0];
            breg[j].y = wp[1];
        }
        float bv = bias ? bias[col < N ? col : 0] : 0.0f;

        for (long long row0 = ((long long)blockIdx.x * GEMM_WAVES + wave) * 16; row0 < n;
             row0 += tileStride) {
            int arow = (int)row0 + l15;
            int arow_c = arow < n ? arow : n - 1;   // clamp: no EXEC churn
            const float* Xr = X + (long long)arow_c * K;
            __builtin_prefetch(X + ((long long)arow_c + tileStride) * K, 0, 1);

            v2f areg[KQ];
#pragma unroll
            for (int j = 0; j < KQ; ++j) {          // batch-issued global_load_b64s
                int k = (j << 2) + (hi << 1);
                areg[j].x = Xr[k];
                areg[j].y = Xr[k + 1];
            }
            v8f acc = {};
#pragma unroll
            for (int j = 0; j < KQ; ++j)
                acc = __builtin_amdgcn_wmma_f32_16x16x4_f32(false, areg[j], false, breg[j],
                                                            (short)0, acc, false, false);

            int rbase = (int)row0 + (hi << 3);
            if ((row0 + 16) <= (long long)n) {      // uniform fast path
                if (col < N) {
#pragma unroll
                    for (int r = 0; r < 8; ++r) {
                        long long o = (long long)(rbase + r) * N + col;
                        float v = acc[r] + bv;
                        if (residual) v += residual[o];
                        if (fuse_gelu) v = gelu_f(v);
                        Y[o] = v;
                    }
                }
            } else if (col < N) {
#pragma unroll
                for (int r = 0; r < 8; ++r) {
                    int row = rbase + r;
                    if (row < n) {
                        long long o = (long long)row * N + col;
                        float v = acc[r] + bv;
                        if (residual) v += residual[o];
                        if (fuse_gelu) v = gelu_f(v);
                        Y[o] = v;
                    }
                }
            }
        }
    }
}

// ---- generic large-K path: software-pipelined, W pairs streamed from LDS.
__global__ __launch_bounds__(256) void gemm16_kernel(const float* __restrict__ X,
                                                     const float* __restrict__ W,
                                                     const float* __restrict__ bias,
                                                     const float* __restrict__ residual,
                                                     float* __restrict__ Y,
                                                     int n, int K, int N, int fuse_gelu) {
    extern __shared__ float smem[];  // (K4+4) * N16 floats, pair-interleaved
    int K4 = (K + 3) & ~3;
    int N16 = (N + 15) & ~15;
    int tid = threadIdx.x;
    int npairs = ((K4 + 4) >> 1) * N16;   // +2 zero pair-rows for the pipeline preload
    for (int i = tid; i < npairs; i += 256) {
        int kp = i / N16;
        int col = i - kp * N16;
        int k = kp << 1;
        float w0 = (k < K && col < N) ? W[k * N + col] : 0.0f;
        float w1 = ((k + 1) < K && col < N) ? W[(k + 1) * N + col] : 0.0f;
        smem[i * 2] = w0;
        smem[i * 2 + 1] = w1;
    }
    __syncthreads();

    int wave = tid >> 5;
    int lane = tid & 31;
    int hi = lane >> 4;
    int l15 = lane & 15;
    int ntiles = N16 >> 4;
    long long tileStride = (long long)gridDim.x * (16 * GEMM_WAVES);

    for (long long row0 = ((long long)blockIdx.x * GEMM_WAVES + wave) * 16; row0 < n;
         row0 += tileStride) {
        int arow = (int)row0 + l15;
        int arow_c = arow < n ? arow : n - 1;
        const float* Xr = X + (long long)arow_c * K;
        __builtin_prefetch(X + ((long long)arow_c + tileStride) * K, 0, 1);
        bool fullTile = (row0 + 16) <= (long long)n;

        for (int t = 0; t < ntiles; ++t) {
            int col = (t << 4) + l15;
            v8f acc = {};
            int k = hi << 1;
            v2f a, bb;
            a.x = Xr[k];
            a.y = Xr[k + 1];
            {
                const float* wp = smem + (((k >> 1) * N16 + col) << 1);
                bb.x = wp[0];
                bb.y = wp[1];
            }
            for (int k0 = 0; k0 < K4; k0 += 4) {
                int kn = k0 + 4 + (hi << 1);      // next-iteration fragment
                v2f an, bn;
                an.x = Xr[kn];
                an.y = Xr[kn + 1];
                const float* wp = smem + (((kn >> 1) * N16 + col) << 1);
                bn.x = wp[0];
                bn.y = wp[1];
                acc = __builtin_amdgcn_wmma_f32_16x16x4_f32(false, a, false, bb,
                                                            (short)0, acc, false, false);
                a = an;
                bb = bn;
            }
            float bv = bias ? bias[col < N ? col : 0] : 0.0f;
            int rbase = (int)row0 + (hi << 3);
            if (fullTile) {
                if (col < N) {
#pragma unroll
                    for (int r = 0; r < 8; ++r) {
                        long long o = (long long)(rbase + r) * N + col;
                        float v = acc[r] + bv;
                        if (residual) v += residual[o];
                        if (fuse_gelu) v = gelu_f(v);
                        Y[o] = v;
                    }
                }
            } else if (col < N) {
#pragma unroll
                for (int r = 0; r < 8; ++r) {
                    int row = rbase + r;
                    if (row < n) {
                        long long o = (long long)row * N + col;
                        float v = acc[r] + bv;
                        if (residual) v += residual[o];
                        if (fuse_gelu) v = gelu_f(v);
                        Y[o] = v;
                    }
                }
            }
        }
    }
}

// -------------------------- graph conv scatter (segment_sum) ---------------
// agg[(row*NET+et), :] += concat(x[col], one_hot(nt[col], T))

__global__ __launch_bounds__(256) void scatter_edges_kernel(const float* __restrict__ x,
                                                            const int* __restrict__ nt,
                                                            const int* __restrict__ erow,
                                                            const int* __restrict__ ecol,
                                                            const int* __restrict__ et,
                                                            long long E, int Cin, int T,
                                                            float* __restrict__ agg) {
    int CT = Cin + T;
    long long e = (long long)blockIdx.x * blockDim.x + threadIdx.x;
    long long st = (long long)gridDim.x * blockDim.x;
    for (; e < E; e += st) {
        int r = erow[e];
        int cl = ecol[e];
        int t = et[e];
        int ty = nt[cl];
        const float* xr = x + (long long)cl * Cin;
        float* ar = agg + ((long long)r * NET + t) * CT;
        for (int c = 0; c < Cin; ++c) atomicAdd(&ar[c], xr[c]);
        atomicAdd(&ar[Cin + ty], 1.0f);
    }
}

// ------------------------------- down/up sample ----------------------------

__global__ __launch_bounds__(256) void down_gather_kernel(const float* __restrict__ x,
                                                          const int* __restrict__ leaf,
                                                          const int* __restrict__ nonleaf,
                                                          int nleaf, int nnl, int numd,
                                                          int lnum, int ntot, int C,
                                                          float* __restrict__ out) {
    long long prefix = (long long)ntot - numd - lnum;
    long long tA = prefix * C;
    long long tB = (long long)nleaf * C;
    long long tC = (long long)nnl * C;
    long long total = tA + tB + tC;
    long long i = (long long)blockIdx.x * blockDim.x + threadIdx.x;
    long long st = (long long)gridDim.x * blockDim.x;
    for (; i < total; i += st) {
        if (i < tA) {
            out[i] = x[i];
        } else if (i < tA + tB) {
            long long j = i - tA;
            long long r = j / C;
            int c = (int)(j - r * C);
            out[(prefix + leaf[r]) * C + c] = x[((long long)ntot - lnum - numd + r) * C + c];
        } else {
            long long j = i - tA - tB;
            long long r = j / C;
            int c = (int)(j - r * C);
            long long base = (long long)ntot - numd + 8 * r;
            float s = 0.0f;
#pragma unroll
            for (int k = 0; k < 8; ++k) s += x[(base + k) * C + c];
            out[(prefix + nonleaf[r]) * C + c] = s * 0.125f;
        }
    }
}

__global__ __launch_bounds__(256) void up_gather_kernel(const float* __restrict__ x,
                                                        const int* __restrict__ leaf,
                                                        const int* __restrict__ nonleaf,
                                                        int nleaf, int nnl, int numd,
                                                        int ntot, int C,
                                                        float* __restrict__ out) {
    long long prefix = (long long)ntot - numd;
    long long tA = prefix * C;
    long long tB = (long long)nleaf * C;
    long long tC = (long long)nnl * 8 * C;
    long long total = tA + tB + tC;
    long long dbase = (long long)ntot - numd;  // start of outd in x
    long long i = (long long)blockIdx.x * blockDim.x + threadIdx.x;
    long long st = (long long)gridDim.x * blockDim.x;
    for (; i < total; i += st) {
        if (i < tA) {
            out[i] = x[i];
        } else if (i < tA + tB) {
            long long j = i - tA;
            long long r = j / C;
            int c = (int)(j - r * C);
            out[(prefix + r) * C + c] = x[(dbase + leaf[r]) * C + c];
        } else {
            long long j = i - tA - tB;
            long long r = j / C;  // 0..nnl*8
            int c = (int)(j - r * C);
            out[(prefix + nleaf + r) * C + c] = x[(dbase + nonleaf[r >> 3]) * C + c];
        }
    }
}

// ------------------------------- KL / latent -------------------------------

__global__ __launch_bounds__(256) void kl_z_kernel(const float* __restrict__ code,
                                                   const float* __restrict__ noise,
                                                   float* __restrict__ z,
                                                   float* __restrict__ klacc, int n) {
    __shared__ float sh;
    if (threadIdx.x == 0) sh = 0.0f;
    __syncthreads();
    int r = blockIdx.x * blockDim.x + threadIdx.x;
    if (r < n) {
        float kl = 0.0f;
#pragma unroll
        for (int j = 0; j < LATC; ++j) {
            float mu = code[r * (2 * LATC) + j];
            float lv = code[r * (2 * LATC) + LATC + j];
            lv = fminf(fmaxf(lv, -30.0f), 20.0f);
            float ev = expf(lv);
            z[r * LATC + j] = mu + expf(0.5f * lv) * noise[r * LATC + j];
            kl += 0.5f * (mu * mu + ev - 1.0f - lv);
        }
        atomicAdd(&sh, kl);
    }
    __syncthreads();
    if (threadIdx.x == 0) atomicAdd(klacc, sh);
}

__global__ void kl_finalize_kernel(const float* __restrict__ acc, float* __restrict__ out,
                                   float inv_n) {
    if (blockIdx.x == 0 && threadIdx.x == 0) out[0] = acc[0] * inv_n;
}

// ------------------------------- host side ---------------------------------

struct BlkP { const float *b1,*b2,*gn1_b,*gn1_w,*gn2_b,*gn2_w,*gn3_b,*gn3_w,*w1,*w2,*wg; };
struct LinP { const float *b, *w; };
struct HeadP { const float *b1,*b2,*gn_b,*gn_w,*w1,*w2; };

static inline dim3 gs_grid(long long total) {
    long long b = (total + 255) / 256;
    if (b < 1) b = 1;
    if (b > 16384) b = 16384;
    return dim3((unsigned)b);
}

extern "C" void kernel_launch(void* const* d_in, const int* in_sizes, int n_in,
                              void* d_out, int out_size, void* d_ws, size_t ws_size,
                              hipStream_t stream) {
    (void)in_sizes; (void)n_in; (void)out_size; (void)ws_size;

    // ---- inputs --------------------------------------------------------
    const float* x_in  = (const float*)d_in[0];
    const float* noise = (const float*)d_in[1];

    // params: jax pytree flatten order (dict keys sorted at each level)
    int ip = 2;
    auto nextf = [&]() { return (const float*)d_in[ip++]; };
    auto loadBlk = [&](BlkP& B) {
        B.b1 = nextf(); B.b2 = nextf();
        B.gn1_b = nextf(); B.gn1_w = nextf();
        B.gn2_b = nextf(); B.gn2_w = nextf();
        B.gn3_b = nextf(); B.gn3_w = nextf();
        B.w1 = nextf(); B.w2 = nextf(); B.wg = nextf();
    };
    auto loadLin = [&](LinP& L) { L.b = nextf(); L.w = nextf(); };
    auto loadHead = [&](HeadP& H) {
        H.b1 = nextf(); H.b2 = nextf(); H.gn_b = nextf(); H.gn_w = nextf();
        H.w1 = nextf(); H.w2 = nextf();
    };

    const float* conv1 = nextf();
    BlkP dec6[3], dec7[3], dec8[3], enc6[2], enc7[2], enc8[2], mid1[3], mid2[3];
    for (int i = 0; i < 3; ++i) loadBlk(dec6[i]);
    for (int i = 0; i < 3; ++i) loadBlk(dec7[i]);
    for (int i = 0; i < 3; ++i) loadBlk(dec8[i]);
    LinP down7, down8, up7, up8;
    loadLin(down7); loadLin(down8);
    for (int i = 0; i < 2; ++i) loadBlk(enc6[i]);
    for (int i = 0; i < 2; ++i) loadBlk(enc7[i]);
    for (int i = 0; i < 2; ++i) loadBlk(enc8[i]);
    const float* enc_gn_b = nextf();
    const float* enc_gn_w = nextf();
    const float* kl_b = nextf();
    const float* kl_w = nextf();
    for (int i = 0; i < 3; ++i) loadBlk(mid1[i]);
    for (int i = 0; i < 3; ++i) loadBlk(mid2[i]);
    const float* post_b = nextf();
    const float* post_w = nextf();
    HeadP pred6, pred7, pred8;
    loadHead(pred6); loadHead(pred7); loadHead(pred8);
    const float* sem_b1 = nextf();
    const float* sem_b2 = nextf();
    const float* sem_w1 = nextf();
    const float* sem_w2 = nextf();
    loadLin(up7); loadLin(up8);

    const int* ei6 = (const int*)d_in[ip++];
    const int* et6 = (const int*)d_in[ip++];
    const int* nt6 = (const int*)d_in[ip++];
    const int* ei7 = (const int*)d_in[ip++];
    const int* et7 = (const int*)d_in[ip++];
    const int* nt7 = (const int*)d_in[ip++];
    const int* ei8 = (const int*)d_in[ip++];
    const int* et8 = (const int*)d_in[ip++];
    const int* nt8 = (const int*)d_in[ip++];
    const int* leaf6 = (const int*)d_in[ip++];
    const int* nonleaf6 = (const int*)d_in[ip++];
    const int* leaf7 = (const int*)d_in[ip++];
    const int* nonleaf7 = (const int*)d_in[ip++];

    // ---- workspace layout (floats) -------------------------------------
    float* ws = (float*)d_ws;
    long long cur = 0;
    auto alloc = [&](long long nfloats) { float* p = ws + cur; cur += nfloats; return p; };
    const long long G8 = G8C;
    float* X   = alloc(G8 * 32);    // canonical activations
    float* X2  = alloc(G8 * 32);    // gather targets
    float* A   = alloc(G8 * 32);    // temp
    float* Bv  = alloc(G8 * 32);    // temp
    float* H64 = alloc(G8 * 64);    // sem hidden
    float* AGG = alloc(G8 * 217);   // graph-conv accumulator (max NET*(24+7))
    float* Z   = alloc((long long)G6C * LATC);
    float* stats = alloc(128);
    float* klacc = alloc(64);       // slack so GEMM over-read stays in-bounds

    float* out = (float*)d_out;
    const long long OFF6 = 0;
    const long long OFF7 = OFF6 + (long long)N6C * 2;
    const long long OFF8 = OFF7 + (long long)N7C * 2;
    const long long OFFS = OFF8 + (long long)N8C * 2;
    const long long OFFK = OFFS + (long long)N8C * NCLSC;

    // ---- helpers -------------------------------------------------------
    auto gemm = [&](const float* Xp, const float* W, const float* bias, const float* res,
                    float* Y, long long n, int K, int N, int gelu_out) {
        long long tb = (n + (16 * GEMM_WAVES) - 1) / (16 * GEMM_WAVES);
        int blocks = tb > 2048 ? 2048 : (int)tb;
        int K4 = (K + 3) & ~3;
        int N16 = (N + 15) & ~15;
        if (K4 <= 32) {
            size_t lds = (size_t)K4 * N16 * sizeof(float);
            dim3 g((unsigned)blocks), blk(256);
            switch (K4 >> 2) {
                case 1: gemm16_reg_kernel<1><<<g, blk, lds, stream>>>(Xp, W, bias, res, Y, (int)n, K, N, gelu_out); break;
                case 2: gemm16_reg_kernel<2><<<g, blk, lds, stream>>>(Xp, W, bias, res, Y, (int)n, K, N, gelu_out); break;
                case 3: gemm16_reg_kernel<3><<<g, blk, lds, stream>>>(Xp, W, bias, res, Y, (int)n, K, N, gelu_out); break;
                case 4: gemm16_reg_kernel<4><<<g, blk, lds, stream>>>(Xp, W, bias, res, Y, (int)n, K, N, gelu_out); break;
                case 5: gemm16_reg_kernel<5><<<g, blk, lds, stream>>>(Xp, W, bias, res, Y, (int)n, K, N, gelu_out); break;
                case 6: gemm16_reg_kernel<6><<<g, blk, lds, stream>>>(Xp, W, bias, res, Y, (int)n, K, N, gelu_out); break;
                case 7: gemm16_reg_kernel<7><<<g, blk, lds, stream>>>(Xp, W, bias, res, Y, (int)n, K, N, gelu_out); break;
                default: gemm16_reg_kernel<8><<<g, blk, lds, stream>>>(Xp, W, bias, res, Y, (int)n, K, N, gelu_out); break;
            }
        } else {
            size_t lds = (size_t)(K4 + 4) * N16 * sizeof(float);
            gemm16_kernel<<<dim3(blocks), dim3(256), lds, stream>>>(Xp, W, bias, res, Y,
                                                                    (int)n, K, N, gelu_out);
        }
    };
    auto groupnorm = [&](const float* Xp, const float* w, const float* b, float* Y,
                         long long n, int C, int do_gelu) {
        zero_kernel<<<1, 128, 0, stream>>>(stats, 128);
        int perB = 256 / C;
        long long sb = (n + perB - 1) / perB;
        if (sb > 4096) sb = 4096;
        colstats_kernel<<<dim3((unsigned)sb), 256, 0, stream>>>(Xp, n, C, stats);
        gn_affine_kernel<<<gs_grid(n * C), 256, 0, stream>>>(Xp, stats, w, b, Y, n, C, do_gelu);
    };
    auto gconv = [&](const float* Xp, const int* ei, const int* et, const int* nt,
                     const float* W, long long n, int Cin, int T, float* Y, int Cout) {
        long long E = (long long)NET * n;
        int CT = Cin + T;
        long long aggN = n * (long long)NET * CT;
        zero_kernel<<<gs_grid(aggN), 256, 0, stream>>>(AGG, aggN);
        scatter_edges_kernel<<<gs_grid(E), 256, 0, stream>>>(Xp, nt, ei, ei + E, et,
                                                             E, Cin, T, AGG);
        gemm(AGG, W, nullptr, nullptr, Y, n, NET * CT, Cout, 0);
    };
    auto resblock = [&](float* xcur, const int* ei, const int* et, const int* nt,
                        const BlkP& B, long long n, int C, int T) {
        int Cb = C / 4;
        groupnorm(xcur, B.gn1_w, B.gn1_b, A, n, C, 1);
        gemm(A, B.w1, B.b1, nullptr, Bv, n, C, Cb, 0);
        groupnorm(Bv, B.gn2_w, B.gn2_b, A, n, Cb, 1);
        gconv(A, ei, et, nt, B.wg, n, Cb, T, Bv, Cb);
        groupnorm(Bv, B.gn3_w, B.gn3_b, A, n, Cb, 1);
        gemm(A, B.w2, B.b2, xcur, xcur, n, Cb, C, 0);  // x + h (read-before-write per elem)
    };
    auto predhead = [&](const float* xcur, const HeadP& P, long long n, int C,
                        long long tailRows, long long off) {
        gemm(xcur, P.w1, P.b1, nullptr, A, n, C, NHC, 0);
        groupnorm(A, P.gn_w, P.gn_b, Bv, n, NHC, 1);
        gemm(Bv, P.w2, P.b2, nullptr, A, n, NHC, 2, 0);
        copy_tail_kernel<<<gs_grid(tailRows * 2), 256, 0, stream>>>(A, n, 2, tailRows,
                                                                    out + off);
    };

    // ---- forward -------------------------------------------------------
    // encoder level 8
    gconv(x_in, ei8, et8, nt8, conv1, G8C, C8C, 7, X, C8C);
    for (int i = 0; i < 2; ++i) resblock(X, ei8, et8, nt8, enc8[i], G8C, C8C, 7);
    // down8: G8 -> G7, C8 -> C7
    down_gather_kernel<<<gs_grid((long long)G7C * C8C), 256, 0, stream>>>(
        X, leaf7, nonleaf7, L7C, S7C, N8C, L7C, G8C, C8C, X2);
    gemm(X2, down8.w, down8.b, nullptr, X, G7C, C8C, C7C, 0);
    // encoder level 7
    for (int i = 0; i < 2; ++i) resblock(X, ei7, et7, nt7, enc7[i], G7C, C7C, 6);
    // down7: G7 -> G6, C7 -> C6
    down_gather_kernel<<<gs_grid((long long)G6C * C7C), 256, 0, stream>>>(
        X, leaf6, nonleaf6, L6C, S6C, N7C, L6C, G7C, C7C, X2);
    gemm(X2, down7.w, down7.b, nullptr, X, G6C, C7C, C6C, 0);
    // encoder level 6
    for (int i = 0; i < 2; ++i) resblock(X, ei6, et6, nt6, enc6[i], G6C, C6C, 5);
    // latent
    groupnorm(X, enc_gn_w, enc_gn_b, A, G6C, C6C, 1);
    gemm(A, kl_w, kl_b, nullptr, Bv, G6C, C6C, 2 * LATC, 0);
    zero_kernel<<<1, 64, 0, stream>>>(klacc, 1);
    kl_z_kernel<<<gs_grid(G6C), 256, 0, stream>>>(Bv, noise, Z, klacc, G6C);
    kl_finalize_kernel<<<1, 32, 0, stream>>>(klacc, out + OFFK, 1.0f / (float)G6C);
    gemm(Z, post_w, post_b, nullptr, X, G6C, LATC, C6C, 0);
    // decoder level 6
    for (int i = 0; i < 3; ++i) resblock(X, ei6, et6, nt6, mid1[i], G6C, C6C, 5);
    for (int i = 0; i < 3; ++i) resblock(X, ei6, et6, nt6, mid2[i], G6C, C6C, 5);
    for (int i = 0; i < 3; ++i) resblock(X, ei6, et6, nt6, dec6[i], G6C, C6C, 5);
    predhead(X, pred6, G6C, C6C, N6C, OFF6);
    // up7: G6 -> G7
    up_gather_kernel<<<gs_grid((long long)G7C * C6C), 256, 0, stream>>>(
        X, leaf6, nonleaf6, L6C, S6C, N6C, G6C, C6C, X2);
    gemm(X2, up7.w, up7.b, nullptr, X, G7C, C6C, C7C, 0);
    for (int i = 0; i < 3; ++i) resblock(X, ei7, et7, nt7, dec7[i], G7C, C7C, 6);
    predhead(X, pred7, G7C, C7C, N7C, OFF7);
    // up8: G7 -> G8
    up_gather_kernel<<<gs_grid((long long)G8C * C7C), 256, 0, stream>>>(
        X, leaf7, nonleaf7, L7C, S7C, N7C, G7C, C7C, X2);
    gemm(X2, up8.w, up8.b, nullptr, X, G8C, C7C, C8C, 0);
    for (int i = 0; i < 3; ++i) resblock(X, ei8, et8, nt8, dec8[i], G8C, C8C, 7);
    predhead(X, pred8, G8C, C8C, N8C, OFF8);
    // semantic head: gelu(x @ w1 + b1) @ w2 + b2, slice last N8 rows
    gemm(X, sem_w1, sem_b1, nullptr, H64, G8C, C8C, 64, 1);
    gemm(H64, sem_w2, sem_b2, nullptr, A, G8C, 64, NCLSC, 0);
    copy_tail_kernel<<<gs_grid((long long)N8C * NCLSC), 256, 0, stream>>>(
        A, G8C, NCLSC, N8C, out + OFFS);
}